// Model_line_veri_38397007626863
// MI455X (gfx1250) — compile-verified
//
#include <hip/hip_runtime.h>
#include <hip/hip_bf16.h>
#include <math.h>

// ---------------------------------------------------------------------------
// Problem constants (match reference)
// ---------------------------------------------------------------------------
#define CCH   128      // channels
#define HH    128
#define WW    128
#define NLINES 8192
#define NPTS   32
#define POOL   4
#define DFC    1024    // hidden dim == d_in == 1024

typedef __attribute__((ext_vector_type(2))) float v2f;
typedef __attribute__((ext_vector_type(8))) float v8f;
typedef __attribute__((ext_vector_type(4))) int   v4i;

// ---------------------------------------------------------------------------
// gfx1250 async global->LDS copy (ASYNCcnt-tracked), with sync fallback.
// Builtin prototype (per hipcc diagnostic): (int4*, int4*, imm offset, imm cpol)
// ---------------------------------------------------------------------------
#if __has_builtin(__builtin_amdgcn_global_load_async_to_lds_b128) && \
    __has_builtin(__builtin_amdgcn_s_wait_asynccnt)
#define HAVE_ASYNC_LDS 1
#else
#define HAVE_ASYNC_LDS 0
#endif

__device__ __forceinline__ void async_copy16(const float* __restrict__ g,
                                             float* __restrict__ l) {
#if HAVE_ASYNC_LDS
  __builtin_amdgcn_global_load_async_to_lds_b128(
      (v4i*)g, (v4i*)l, /*offset=*/0, /*cpol=*/0);
#else
  *(float4*)l = *(const float4*)g;
#endif
}

template <int N>
__device__ __forceinline__ void wait_async() {
#if HAVE_ASYNC_LDS
  __builtin_amdgcn_s_wait_asynccnt(N);
#endif
}

// ---------------------------------------------------------------------------
// 1) Transpose features (C,H,W) -> (H,W,C) so per-point channel gathers are
//    contiguous 512B reads.
// ---------------------------------------------------------------------------
__global__ __launch_bounds__(1024) void transpose_feat(const float* __restrict__ F,
                                                       float* __restrict__ T) {
  __shared__ float tile[32][33];
  const int h  = blockIdx.x;
  const int wb = blockIdx.y * 32;
  const int cb = blockIdx.z * 32;
  const int tx = threadIdx.x, ty = threadIdx.y;
  tile[ty][tx] = F[((size_t)(cb + ty) * HH + h) * WW + wb + tx];
  __syncthreads();
  T[((size_t)h * WW + (wb + ty)) * CCH + cb + tx] = tile[tx][ty];
}

// ---------------------------------------------------------------------------
// 2) Bilinear line sampling + max-pool(4) -> flat (NLINES x 1024)
//    One block per line, one thread per channel (coalesced channel gathers).
// ---------------------------------------------------------------------------
__global__ __launch_bounds__(128) void sample_pool(const float* __restrict__ featT,
                                                   const float* __restrict__ lines,
                                                   float* __restrict__ flat) {
  const int l = blockIdx.x;
  const int c = threadIdx.x;
  const float U0 = lines[l * 4 + 0];
  const float U1 = lines[l * 4 + 1];
  const float V0 = lines[l * 4 + 2];
  const float V1 = lines[l * 4 + 3];
  float* __restrict__ orow = flat + (size_t)l * DFC + (size_t)c * (NPTS / POOL);

  #pragma unroll
  for (int q = 0; q < NPTS / POOL; ++q) {
    float m = -INFINITY;
    #pragma unroll
    for (int s = 0; s < POOL; ++s) {
      const int   p = q * POOL + s;
      const float t = (float)p * (1.0f / (float)(NPTS - 1));
      const float px = U0 * t + V0 * (1.0f - t) - 0.5f;   // indexes H
      const float py = U1 * t + V1 * (1.0f - t) - 0.5f;   // indexes W
      const float px0 = fminf(fmaxf(floorf(px), 0.0f), (float)(WW - 1));
      const float py0 = fminf(fmaxf(floorf(py), 0.0f), (float)(HH - 1));
      const float px1 = fminf(px0 + 1.0f, (float)(WW - 1));
      const float py1 = fminf(py0 + 1.0f, (float)(HH - 1));
      const int ix0 = (int)px0, iy0 = (int)py0;
      const int ix1 = (int)px1, iy1 = (int)py1;
      const float w00 = (py1 - py) * (px1 - px);
      const float w01 = (py - py0) * (px1 - px);
      const float w10 = (py1 - py) * (px - px0);
      const float w11 = (py - py0) * (px - px0);
      const float f00 = featT[((size_t)ix0 * WW + iy0) * CCH + c];
      const float f01 = featT[((size_t)ix0 * WW + iy1) * CCH + c];
      const float f10 = featT[((size_t)ix1 * WW + iy0) * CCH + c];
      const float f11 = featT[((size_t)ix1 * WW + iy1) * CCH + c];
      const float v = f00 * w00 + f01 * w01 + f10 * w10 + f11 * w11;
      m = fmaxf(m, v);
    }
    orow[q] = m;
  }
}

// ---------------------------------------------------------------------------
// 3) fp32 WMMA GEMM:  Y[M,N] = act(X[M,K] @ Wt[N,K]^T + bias[N])
//    256 threads (8 waves), tile 128(M) x 64(N), K-chunk 32.
//    Double-buffered LDS filled by GLOBAL_LOAD_ASYNC_TO_LDS_B128:
//    chunk i+1's copies are in flight while chunk i's 32 WMMAs per wave execute.
//    V_WMMA_F32_16X16X4_F32 keeps full fp32 precision (matches reference).
// ---------------------------------------------------------------------------
#define LPAD 40   // 160B rows: 16B aligned for B128 LDS writes; 60KB total LDS

template <bool RELU>
__global__ __launch_bounds__(256) void gemm_wmma_f32(const float* __restrict__ X,
                                                     const float* __restrict__ Wt,
                                                     const float* __restrict__ bias,
                                                     float* __restrict__ Y,
                                                     int M, int N, int K) {
  __shared__ float lX[2][128][LPAD];
  __shared__ float lW[2][64][LPAD];

  const int tid  = threadIdx.x;
  const int lane = tid & 31;
  const int wave = tid >> 5;
  const int m0 = blockIdx.x * 128;
  const int n0 = blockIdx.y * 64;

  v8f acc[4] = {v8f{}, v8f{}, v8f{}, v8f{}};

  const int l15   = lane & 15;
  const int klane = (lane >> 4) << 1;   // 0 or 2 : K-pair selector (ISA A/B layout)
  const int mrow  = wave * 16 + l15;

  // issue async copies for one K-chunk into buffer `buf` (6 x b128 per thread)
  auto issue = [&](int kc, int buf) {
    #pragma unroll
    for (int i = 0; i < 4; ++i) {             // X tile: 128 rows x 32 cols
      const int idx = tid + i * 256;
      const int r = idx >> 3, kq = idx & 7;
      async_copy16(X + (size_t)(m0 + r) * K + kc + kq * 4, &lX[buf][r][kq * 4]);
    }
    #pragma unroll
    for (int i = 0; i < 2; ++i) {             // W tile: 64 rows x 32 cols
      const int idx = tid + i * 256;
      const int r = idx >> 3, kq = idx & 7;
      async_copy16(Wt + (size_t)(n0 + r) * K + kc + kq * 4, &lW[buf][r][kq * 4]);
    }
  };

  const int nch = K >> 5;                      // K/32 chunks
  issue(0, 0);

  for (int i = 0; i < nch; ++i) {
    const int buf = i & 1;
    if (i + 1 < nch) {
      issue((i + 1) << 5, buf ^ 1);
      wait_async<6>();   // >=6 of 12 outstanding done (in-order) => chunk i landed
    } else {
      wait_async<0>();
    }
    __syncthreads();     // cross-wave visibility of this chunk's LDS tile

    #pragma unroll
    for (int k = 0; k < 32; k += 4) {
      const v2f a = *(const v2f*)&lX[buf][mrow][k + klane];          // A 16x4 frag
      #pragma unroll
      for (int t = 0; t < 4; ++t) {
        const v2f b = *(const v2f*)&lW[buf][t * 16 + l15][k + klane]; // B 4x16 frag
        acc[t] = __builtin_amdgcn_wmma_f32_16x16x4_f32(
            /*neg_a=*/false, a, /*neg_b=*/false, b,
            /*c_mod=*/(short)0, acc[t], /*reuse_a=*/false, /*reuse_b=*/false);
      }
    }
    __syncthreads();     // protect buf from being refilled while still in use
  }

  // epilogue: bias (+ReLU) + store. C/D layout: VGPR r -> M=r (lanes 0-15)
  // or M=r+8 (lanes 16-31); N = lane&15.
  const int mbase = m0 + wave * 16 + ((lane >> 4) << 3);
  #pragma unroll
  for (int t = 0; t < 4; ++t) {
    const int col = n0 + t * 16 + l15;
    const float bv = bias[col];
    #pragma unroll
    for (int r = 0; r < 8; ++r) {
      float v = acc[t][r] + bv;
      if (RELU) v = fmaxf(v, 0.0f);
      Y[(size_t)(mbase + r) * N + col] = v;
    }
  }
}

// ---------------------------------------------------------------------------
// 4) Head: out[n, j] = h2[n,:] . W3[j,:] + b3[j],  j < 4.
// ---------------------------------------------------------------------------
__global__ __launch_bounds__(128) void head_gemm(const float* __restrict__ H2,
                                                 const float* __restrict__ W3,
                                                 const float* __restrict__ b3,
                                                 float* __restrict__ out) {
  const int n    = blockIdx.x;
  const int j    = threadIdx.x >> 5;
  const int lane = threadIdx.x & 31;
  const float* __restrict__ hrow = H2 + (size_t)n * DFC;
  const float* __restrict__ wrow = W3 + (size_t)j * DFC;
  float s = 0.0f;
  for (int k = lane * 4; k < DFC; k += 32 * 4) {
    const float4 hv = *(const float4*)(hrow + k);
    const float4 wv = *(const float4*)(wrow + k);
    s += hv.x * wv.x + hv.y * wv.y + hv.z * wv.z + hv.w * wv.w;
  }
  #pragma unroll
  for (int off = 16; off > 0; off >>= 1) s += __shfl_down(s, off, 32);
  if (lane == 0) out[n * 4 + j] = s + b3[j];
}

// ---------------------------------------------------------------------------
// Launcher
// ---------------------------------------------------------------------------
extern "C" void kernel_launch(void* const* d_in, const int* in_sizes, int n_in,
                              void* d_out, int out_size, void* d_ws, size_t ws_size,
                              hipStream_t stream) {
  const float* feats = (const float*)d_in[0];   // (128,128,128)
  const float* lines = (const float*)d_in[1];   // (8192,4)
  const float* W1    = (const float*)d_in[2];   // (1024,1024)
  const float* b1    = (const float*)d_in[3];
  const float* W2    = (const float*)d_in[4];   // (1024,1024)
  const float* b2    = (const float*)d_in[5];
  const float* W3    = (const float*)d_in[6];   // (4,1024)
  const float* b3    = (const float*)d_in[7];
  float* out = (float*)d_out;                   // (8192,4)

  char* ws = (char*)d_ws;
  float* featT = (float*)ws;                                    //  8 MB (H,W,C)
  float* flat  = (float*)(ws + (size_t)(8u << 20));             // 32 MB (also h2)
  float* h1    = (float*)(ws + (size_t)(8u << 20) + (size_t)(32u << 20)); // 32 MB

  transpose_feat<<<dim3(HH, WW / 32, CCH / 32), dim3(32, 32), 0, stream>>>(feats, featT);
  sample_pool<<<NLINES, CCH, 0, stream>>>(featT, lines, flat);

  gemm_wmma_f32<true><<<dim3(NLINES / 128, DFC / 64), 256, 0, stream>>>(
      flat, W1, b1, h1, NLINES, DFC, DFC);
  gemm_wmma_f32<true><<<dim3(NLINES / 128, DFC / 64), 256, 0, stream>>>(
      h1, W2, b2, flat /* reuse as h2 */, NLINES, DFC, DFC);

  head_gemm<<<NLINES, 128, 0, stream>>>(flat, W3, b3, out);
}